// Mamba3Block_62010737819992
// MI455X (gfx1250) — compile-verified
//
#include <hip/hip_runtime.h>
#include <math.h>

// ---------------- problem constants ----------------
#define D_MODEL 1024
#define E_DIM   2048
#define N_STATE 128
#define TD      64
#define B_SZ    4
#define L_SEQ   2048
#define M_TOT   (B_SZ * L_SEQ)      // 8192 rows
#define Z_DIM   4419                 // E+E+N+N+1+1+TD+1
#define ZP      4480                 // padded (35 * 128)
#define EPSV    1e-5f
// z column offsets
#define OFF_X    0
#define OFF_GATE 2048
#define OFF_B    4096
#define OFF_C    4224
#define OFF_DT   4352
#define OFF_A    4353
#define OFF_TH   4354
#define OFF_LAM  4418

typedef float v2f __attribute__((ext_vector_type(2)));
typedef float v8f __attribute__((ext_vector_type(8)));

__device__ __forceinline__ float siluf(float v)     { return v / (1.0f + __expf(-v)); }
__device__ __forceinline__ float softplusf(float x) { return (x > 20.0f) ? x : log1pf(__expf(x)); }

// ---------------- weight pack: Wp[((k/4)*NP + n)*4 + (k&3)] = W[n*cols + k], zero-padded n ----------------
__global__ __launch_bounds__(256) void pack_bk4_k(
    const float* __restrict__ W, float* __restrict__ Wp,
    int rows, int cols, int NP) {
  int k = blockIdx.x * 32 + threadIdx.x;   // coalesced read along K
  int n = blockIdx.y * 8 + threadIdx.y;
  if (k >= cols || n >= NP) return;
  float v = (n < rows) ? W[(size_t)n * cols + k] : 0.0f;
  Wp[((size_t)(k >> 2) * NP + n) * 4 + (k & 3)] = v;
}

// ---------------- GEMM1: Z[M_TOT x ZP] = U[M_TOT x 1024] * W_in^T ----------------
// WG: 8 waves, tile 128(M) x 128(N). Each wave: 32(M) x 64(N) = 8 WMMA tiles / k-step.
__global__ __launch_bounds__(256) void gemm1_wmma_k(
    const float* __restrict__ U, const float* __restrict__ Wp,
    float* __restrict__ Z) {
  const int K = D_MODEL, NP = ZP;
  int w    = threadIdx.x >> 5;
  int lane = threadIdx.x & 31;
  int lr   = lane & 15;
  int half = lane >> 4;
  int mbase = blockIdx.y * 128 + (w & 3) * 32;
  int nbase = blockIdx.x * 128 + (w >> 2) * 64;

  const float* arow0 = U + (size_t)(mbase + lr) * K + 2 * half;
  const float* arow1 = arow0 + (size_t)16 * K;
  const float* bbase = Wp + ((size_t)(nbase + lr)) * 4 + 2 * half;

  v8f c00 = {}, c01 = {}, c02 = {}, c03 = {};
  v8f c10 = {}, c11 = {}, c12 = {}, c13 = {};

  for (int kk = 0; kk < K; kk += 4) {
    v2f a0 = *(const v2f*)(arow0 + kk);
    v2f a1 = *(const v2f*)(arow1 + kk);
    const float* bp = bbase + (size_t)(kk >> 2) * NP * 4;
    v2f b0 = *(const v2f*)(bp + 0);
    v2f b1 = *(const v2f*)(bp + 64);
    v2f b2 = *(const v2f*)(bp + 128);
    v2f b3 = *(const v2f*)(bp + 192);
    // unguarded speculative prefetch 8 k-groups ahead (overshoot stays in ws; dropped if invalid)
    __builtin_prefetch(bp + (size_t)8 * NP * 4, 0, 1);
    c00 = __builtin_amdgcn_wmma_f32_16x16x4_f32(false, a0, false, b0, (short)0, c00, false, false);
    c01 = __builtin_amdgcn_wmma_f32_16x16x4_f32(false, a0, false, b1, (short)0, c01, false, false);
    c02 = __builtin_amdgcn_wmma_f32_16x16x4_f32(false, a0, false, b2, (short)0, c02, false, false);
    c03 = __builtin_amdgcn_wmma_f32_16x16x4_f32(false, a0, false, b3, (short)0, c03, false, false);
    c10 = __builtin_amdgcn_wmma_f32_16x16x4_f32(false, a1, false, b0, (short)0, c10, false, false);
    c11 = __builtin_amdgcn_wmma_f32_16x16x4_f32(false, a1, false, b1, (short)0, c11, false, false);
    c12 = __builtin_amdgcn_wmma_f32_16x16x4_f32(false, a1, false, b2, (short)0, c12, false, false);
    c13 = __builtin_amdgcn_wmma_f32_16x16x4_f32(false, a1, false, b3, (short)0, c13, false, false);
  }
  float* zr0 = Z + (size_t)(mbase + 8 * half) * ZP + nbase + lr;
  float* zr1 = zr0 + (size_t)16 * ZP;
  #pragma unroll
  for (int r = 0; r < 8; ++r) {
    zr0[(size_t)r * ZP + 0]  = c00[r];
    zr0[(size_t)r * ZP + 16] = c01[r];
    zr0[(size_t)r * ZP + 32] = c02[r];
    zr0[(size_t)r * ZP + 48] = c03[r];
    zr1[(size_t)r * ZP + 0]  = c10[r];
    zr1[(size_t)r * ZP + 16] = c11[r];
    zr1[(size_t)r * ZP + 32] = c12[r];
    zr1[(size_t)r * ZP + 48] = c13[r];
  }
}

// ---------------- per-token scalars + xmean ----------------
__global__ __launch_bounds__(256) void p1_token_k(
    const float* __restrict__ Z, const float* __restrict__ dt_bias,
    const float* __restrict__ A_log,
    float* __restrict__ dt, float* __restrict__ alpha, float* __restrict__ lam,
    float* __restrict__ xmean, float* __restrict__ Aval) {
  int t = blockIdx.x;
  const float* zr = Z + (size_t)t * ZP;
  float s = 0.0f;
  for (int k = threadIdx.x; k < E_DIM; k += 256) s += siluf(zr[OFF_X + k]);
  __shared__ float red[256];
  red[threadIdx.x] = s; __syncthreads();
  for (int o = 128; o; o >>= 1) { if (threadIdx.x < o) red[threadIdx.x] += red[threadIdx.x + o]; __syncthreads(); }
  if (threadIdx.x == 0) {
    xmean[t] = red[0] * (1.0f / (float)E_DIM);
    float dtv = softplusf(zr[OFF_DT] + dt_bias[0]);
    float Av  = -softplusf(zr[OFF_A] + A_log[0]);
    dt[t]    = dtv;
    alpha[t] = __expf(dtv * Av);
    lam[t]   = 1.0f / (1.0f + __expf(-zr[OFF_LAM]));
    Aval[t]  = Av;
  }
}

// ---------------- deterministic mean of Aval -> Abar ----------------
__global__ __launch_bounds__(256) void reduceA_k(const float* __restrict__ Aval, float* __restrict__ Abar) {
  float s = 0.0f;
  for (int i = threadIdx.x; i < M_TOT; i += 256) s += Aval[i];
  __shared__ float red[256];
  red[threadIdx.x] = s; __syncthreads();
  for (int o = 128; o; o >>= 1) { if (threadIdx.x < o) red[threadIdx.x] += red[threadIdx.x + o]; __syncthreads(); }
  if (threadIdx.x == 0) Abar[0] = red[0] * (1.0f / (float)M_TOT);
}

// ---------------- RoPE angle cumsum: ca[t*TD + d] ----------------
__global__ __launch_bounds__(64) void cumsum_k(
    const float* __restrict__ Z, const float* __restrict__ dt, float* __restrict__ ca) {
  int b = blockIdx.x, d = threadIdx.x;
  float acc = 0.0f;
  for (int l = 0; l < L_SEQ; ++l) {
    int t = b * L_SEQ + l;
    acc += dt[t] * Z[(size_t)t * ZP + OFF_TH + d];
    ca[(size_t)t * TD + d] = acc;
  }
}

// ---------------- RMS-norm + bias + RoPE for B_, C_ ----------------
__global__ __launch_bounds__(128) void bc_k(
    const float* __restrict__ Z, const float* __restrict__ wB, const float* __restrict__ wC,
    const float* __restrict__ Bbias, const float* __restrict__ Cbias,
    const float* __restrict__ ca, float* __restrict__ Bo, float* __restrict__ Co) {
  int t = blockIdx.x, n = threadIdx.x;
  const float* zr = Z + (size_t)t * ZP;
  float braw = zr[OFF_B + n];
  float craw = zr[OFF_C + n];
  __shared__ float red[128];
  __shared__ float sB[128], sC[128];

  red[n] = braw * braw; __syncthreads();
  for (int o = 64; o; o >>= 1) { if (n < o) red[n] += red[n + o]; __syncthreads(); }
  float invB = rsqrtf(red[0] * (1.0f / (float)N_STATE) + EPSV);
  __syncthreads();
  red[n] = craw * craw; __syncthreads();
  for (int o = 64; o; o >>= 1) { if (n < o) red[n] += red[n + o]; __syncthreads(); }
  float invC = rsqrtf(red[0] * (1.0f / (float)N_STATE) + EPSV);
  __syncthreads();

  sB[n] = braw * invB * wB[n] + Bbias[n];
  sC[n] = craw * invC * wC[n] + Cbias[n];
  __syncthreads();

  int d = n & 63;
  float ang = ca[(size_t)t * TD + d];
  float sv, cv; sincosf(ang, &sv, &cv);
  float b1 = sB[d], b2 = sB[d + 64];
  float c1 = sC[d], c2 = sC[d + 64];
  Bo[(size_t)t * N_STATE + n] = (n < 64) ? (b1 * cv - b2 * sv) : (b1 * sv + b2 * cv);
  Co[(size_t)t * N_STATE + n] = (n < 64) ? (c1 * cv - c2 * sv) : (c1 * sv + c2 * cv);
}

// ---------------- recurrent scan: one wave per batch, 4 states/lane ----------------
__global__ __launch_bounds__(32) void scan_k(
    const float* __restrict__ Bm, const float* __restrict__ Cm,
    const float* __restrict__ dt, const float* __restrict__ alpha,
    const float* __restrict__ lam, const float* __restrict__ xmean,
    const float* __restrict__ Abar, float* __restrict__ ys) {
  int b = blockIdx.x, lane = threadIdx.x;
  float Ab = Abar[0];
  float st0 = 0.f, st1 = 0.f, st2 = 0.f, st3 = 0.f;
  float xm_prev = 0.f;
  for (int l = 0; l < L_SEQ; ++l) {
    int t = b * L_SEQ + l;
    float dtv = dt[t], lm = lam[t], al = alpha[t], xm = xmean[t];
    float um = lm * xm + (1.0f - lm) * al * xm_prev;
    xm_prev = xm;
    float dA = __expf(dtv * Ab);
    float du = dtv * um;
    const float* bp = Bm + (size_t)t * N_STATE + lane;
    const float* cp = Cm + (size_t)t * N_STATE + lane;
    st0 = dA * st0 + du * bp[0];  float acc = st0 * cp[0];
    st1 = dA * st1 + du * bp[32]; acc += st1 * cp[32];
    st2 = dA * st2 + du * bp[64]; acc += st2 * cp[64];
    st3 = dA * st3 + du * bp[96]; acc += st3 * cp[96];
    #pragma unroll
    for (int o = 16; o; o >>= 1) acc += __shfl_xor(acc, o, 32);
    if (lane == 0) ys[t] = acc;
  }
}

// ---------------- GEMM2: Out[M_TOT x 1024] = (ys*silu(gate)) @ W_out^T ----------------
__global__ __launch_bounds__(256) void gemm2_wmma_k(
    const float* __restrict__ Z, const float* __restrict__ Wp,
    const float* __restrict__ ys, float* __restrict__ Out) {
  const int K = E_DIM, NP = D_MODEL;
  int w    = threadIdx.x >> 5;
  int lane = threadIdx.x & 31;
  int lr   = lane & 15;
  int half = lane >> 4;
  int mbase = blockIdx.y * 128 + (w & 3) * 32;
  int nbase = blockIdx.x * 128 + (w >> 2) * 64;

  int arow0i = mbase + lr;
  int arow1i = arow0i + 16;
  const float* zr0 = Z + (size_t)arow0i * ZP + OFF_GATE + 2 * half;
  const float* zr1 = Z + (size_t)arow1i * ZP + OFF_GATE + 2 * half;
  float scale0 = ys[arow0i];
  float scale1 = ys[arow1i];
  const float* bbase = Wp + ((size_t)(nbase + lr)) * 4 + 2 * half;

  v8f c00 = {}, c01 = {}, c02 = {}, c03 = {};
  v8f c10 = {}, c11 = {}, c12 = {}, c13 = {};

  for (int kk = 0; kk < K; kk += 4) {
    v2f g0 = *(const v2f*)(zr0 + kk);
    v2f g1 = *(const v2f*)(zr1 + kk);
    v2f a0, a1;
    a0.x = siluf(g0.x) * scale0; a0.y = siluf(g0.y) * scale0;
    a1.x = siluf(g1.x) * scale1; a1.y = siluf(g1.y) * scale1;
    const float* bp = bbase + (size_t)(kk >> 2) * NP * 4;
    v2f b0 = *(const v2f*)(bp + 0);
    v2f b1 = *(const v2f*)(bp + 64);
    v2f b2 = *(const v2f*)(bp + 128);
    v2f b3 = *(const v2f*)(bp + 192);
    __builtin_prefetch(bp + (size_t)8 * NP * 4, 0, 1);
    c00 = __builtin_amdgcn_wmma_f32_16x16x4_f32(false, a0, false, b0, (short)0, c00, false, false);
    c01 = __builtin_amdgcn_wmma_f32_16x16x4_f32(false, a0, false, b1, (short)0, c01, false, false);
    c02 = __builtin_amdgcn_wmma_f32_16x16x4_f32(false, a0, false, b2, (short)0, c02, false, false);
    c03 = __builtin_amdgcn_wmma_f32_16x16x4_f32(false, a0, false, b3, (short)0, c03, false, false);
    c10 = __builtin_amdgcn_wmma_f32_16x16x4_f32(false, a1, false, b0, (short)0, c10, false, false);
    c11 = __builtin_amdgcn_wmma_f32_16x16x4_f32(false, a1, false, b1, (short)0, c11, false, false);
    c12 = __builtin_amdgcn_wmma_f32_16x16x4_f32(false, a1, false, b2, (short)0, c12, false, false);
    c13 = __builtin_amdgcn_wmma_f32_16x16x4_f32(false, a1, false, b3, (short)0, c13, false, false);
  }
  float* or0 = Out + (size_t)(mbase + 8 * half) * NP + nbase + lr;
  float* or1 = or0 + (size_t)16 * NP;
  #pragma unroll
  for (int r = 0; r < 8; ++r) {
    or0[(size_t)r * NP + 0]  = c00[r];
    or0[(size_t)r * NP + 16] = c01[r];
    or0[(size_t)r * NP + 32] = c02[r];
    or0[(size_t)r * NP + 48] = c03[r];
    or1[(size_t)r * NP + 0]  = c10[r];
    or1[(size_t)r * NP + 16] = c11[r];
    or1[(size_t)r * NP + 32] = c12[r];
    or1[(size_t)r * NP + 48] = c13[r];
  }
}

// ---------------- host launcher ----------------
extern "C" void kernel_launch(void* const* d_in, const int* in_sizes, int n_in,
                              void* d_out, int out_size, void* d_ws, size_t ws_size,
                              hipStream_t stream) {
  const float* u     = (const float*)d_in[0];
  const float* W_in  = (const float*)d_in[1];
  const float* wNB   = (const float*)d_in[2];
  const float* wNC   = (const float*)d_in[3];
  const float* Bbias = (const float*)d_in[4];
  const float* Cbias = (const float*)d_in[5];
  const float* dtb   = (const float*)d_in[6];
  const float* Alog  = (const float*)d_in[7];
  const float* W_out = (const float*)d_in[8];
  float* out = (float*)d_out;
  float* ws  = (float*)d_ws;

  // workspace layout (floats, 64-elem aligned)
  size_t off = 0;
  auto take = [&](size_t n) { size_t o = off; off += (n + 63) & ~(size_t)63; return o; };
  float* Z     = ws + take((size_t)M_TOT * ZP);        // 8192 x 4480
  float* Wp1   = ws + take((size_t)D_MODEL * ZP);      // packed (1024/4) x 4480 x 4
  float* Wp2   = ws + take((size_t)E_DIM * D_MODEL);   // packed (2048/4) x 1024 x 4
  float* dtA   = ws + take(M_TOT);
  float* alpha = ws + take(M_TOT);
  float* lamA  = ws + take(M_TOT);
  float* xmean = ws + take(M_TOT);
  float* Aval  = ws + take(M_TOT);
  float* Abar  = ws + take(64);
  float* ca    = ws + take((size_t)M_TOT * TD);
  float* Bm    = ws + take((size_t)M_TOT * N_STATE);
  float* Cm    = ws + take((size_t)M_TOT * N_STATE);
  float* ysA   = ws + take(M_TOT);
  (void)ws_size; (void)in_sizes; (void)n_in; (void)out_size;

  // 1) pack weights (k-major groups of 4, zero-padded n)
  pack_bk4_k<<<dim3(D_MODEL / 32, ZP / 8), dim3(32, 8), 0, stream>>>(
      W_in, Wp1, Z_DIM, D_MODEL, ZP);
  pack_bk4_k<<<dim3(E_DIM / 32, D_MODEL / 8), dim3(32, 8), 0, stream>>>(
      W_out, Wp2, D_MODEL, E_DIM, D_MODEL);

  // 2) GEMM1: Z = u @ W_in^T
  gemm1_wmma_k<<<dim3(ZP / 128, M_TOT / 128), 256, 0, stream>>>(u, Wp1, Z);

  // 3) per-token scalars + xmean
  p1_token_k<<<M_TOT, 256, 0, stream>>>(Z, dtb, Alog, dtA, alpha, lamA, xmean, Aval);

  // 4) A_bar (deterministic tree reduce)
  reduceA_k<<<1, 256, 0, stream>>>(Aval, Abar);

  // 5) RoPE angle cumsum
  cumsum_k<<<B_SZ, TD, 0, stream>>>(Z, dtA, ca);

  // 6) B_/C_ : RMS + bias + RoPE
  bc_k<<<M_TOT, N_STATE, 0, stream>>>(Z, wNB, wNC, Bbias, Cbias, ca, Bm, Cm);

  // 7) recurrent scan -> ys
  scan_k<<<B_SZ, 32, 0, stream>>>(Bm, Cm, dtA, alpha, lamA, xmean, Abar, ysA);

  // 8) GEMM2: out = (ys * silu(gate)) @ W_out^T
  gemm2_wmma_k<<<dim3(D_MODEL / 128, M_TOT / 128), 256, 0, stream>>>(Z, Wp2, ysA, out);
}